// SparseContrastiveAutoEncoder_71090298684114
// MI455X (gfx1250) — compile-verified
//
#include <hip/hip_runtime.h>

typedef __attribute__((ext_vector_type(2))) float v2f;
typedef __attribute__((ext_vector_type(8))) float v8f;

#define BM 128
#define BN 128
#define BK 32
#define LDK 36        // padded LDS row stride (floats); 36*4=144B, multiple of 16B
#define LISTCAP 512   // max k among the three codes

// ---------------------------------------------------------------------------
// Kernel 0: center the input once: xc[b,d] = x[b,d] - bias[d].
// Removes all bias math (and its load-use bubbles) from the WMMA hot loop.
// ---------------------------------------------------------------------------
__global__ __launch_bounds__(256)
void center_x(const float* __restrict__ x, const float* __restrict__ bias,
              float* __restrict__ xc, int D, long long n4)
{
    const long long i = (long long)blockIdx.x * 256 + threadIdx.x;
    if (i >= n4) return;
    const int d4 = (int)(i % (D >> 2));
    float4 v = ((const float4*)x)[i];
    const float4 b = ((const float4*)bias)[d4];
    v.x -= b.x; v.y -= b.y; v.z -= b.z; v.w -= b.w;
    ((float4*)xc)[i] = v;
}

// ---------------------------------------------------------------------------
// Kernel 1: transpose W_dec [D,H] -> W_decT [H,D] so decoder gathers are
// contiguous rows (coalesced b128 loads).
// ---------------------------------------------------------------------------
__global__ __launch_bounds__(256)
void transpose_wdec(const float* __restrict__ Wd, float* __restrict__ WdT,
                    int D, int H)
{
    __shared__ float t[32][33];
    const int bh = blockIdx.x * 32;
    const int bd = blockIdx.y * 32;
    const int tx = threadIdx.x & 31;
    const int ty = threadIdx.x >> 5;   // 0..7
#pragma unroll
    for (int i = 0; i < 4; ++i) {
        const int d = bd + ty + i * 8;
        t[ty + i * 8][tx] = Wd[(size_t)d * H + bh + tx];
    }
    __syncthreads();
#pragma unroll
    for (int i = 0; i < 4; ++i) {
        const int h = bh + ty + i * 8;
        WdT[(size_t)h * D + bd + tx] = t[tx][ty + i * 8];
    }
}

// ---------------------------------------------------------------------------
// Kernel 2: encoder GEMM  pre[b,h] = sum_d xc[b,d] * W_enc[h,d]
// 128x128 tile per 256-thread block (8 waves); V_WMMA_F32_16X16X4_F32.
// Double-buffered LDS staging via GLOBAL_LOAD_ASYNC_TO_LDS_B128 (ASYNCcnt):
// chunk i+1 streams into the other LDS buffer while chunk i feeds WMMAs.
// Inner loop is pure ds_load -> v_wmma (no global loads on the critical path).
// ---------------------------------------------------------------------------
__global__ __launch_bounds__(256)
void enc_gemm(const float* __restrict__ xc, const float* __restrict__ W,
              float* __restrict__ pre, int D, int H)
{
    __shared__ float xs[2][BM * LDK];
    __shared__ float wsh[2][BN * LDK];

    const int h0   = blockIdx.x * BN;
    const int b0   = blockIdx.y * BM;
    const int tid  = threadIdx.x;
    const int lane = tid & 31;
    const int wave = tid >> 5;
    const int lo   = lane & 15;
    const int koff = (lane >= 16) ? 2 : 0;

    const unsigned long long xbase = (unsigned long long)(uintptr_t)xc;
    const unsigned long long wbase = (unsigned long long)(uintptr_t)W;

    // issue 8 async 16B copies per thread: 128x32 floats of xc and of W_enc
    auto issue = [&](int buf, int kc) {
#pragma unroll
        for (int i = 0; i < 4; ++i) {
            const int id  = tid + i * 256;   // float4 chunk id, 0..1023
            const int row = id >> 3;         // 0..127
            const int c4  = (id & 7) << 2;   // 0..28 step 4
            const unsigned lx = (unsigned)(uintptr_t)&xs[buf][row * LDK + c4];
            const unsigned ox = (unsigned)(((size_t)(b0 + row) * D + kc + c4)
                                           * sizeof(float));
            asm volatile("global_load_async_to_lds_b128 %0, %1, %2"
                         :: "v"(lx), "v"(ox), "s"(xbase) : "memory");
            const unsigned lw = (unsigned)(uintptr_t)&wsh[buf][row * LDK + c4];
            const unsigned ow = (unsigned)(((size_t)(h0 + row) * D + kc + c4)
                                           * sizeof(float));
            asm volatile("global_load_async_to_lds_b128 %0, %1, %2"
                         :: "v"(lw), "v"(ow), "s"(wbase) : "memory");
        }
    };

    v8f acc[8] = {};

    const int nch = D / BK;
    issue(0, 0);

    for (int kci = 0; kci < nch; ++kci) {
        const int buf = kci & 1;

        if (kci + 1 < nch) {
            issue(buf ^ 1, (kci + 1) * BK);       // prefetch next chunk
            // our 8 oldest async ops (this chunk) done; next chunk in flight
            asm volatile("s_wait_asynccnt 0x8" ::: "memory");
        } else {
            asm volatile("s_wait_asynccnt 0x0" ::: "memory");
        }
        __syncthreads();                          // all waves' tiles visible

#pragma unroll
        for (int k = 0; k < BK; k += 4) {
            const v2f a =
                *(const v2f*)(&xs[buf][(wave * 16 + lo) * LDK + k + koff]);
#pragma unroll
            for (int j = 0; j < 8; ++j) {
                const v2f bf =
                    *(const v2f*)(&wsh[buf][(j * 16 + lo) * LDK + k + koff]);
                acc[j] = __builtin_amdgcn_wmma_f32_16x16x4_f32(
                    /*neg_a=*/false, a, /*neg_b=*/false, bf,
                    /*c_mod=*/(short)0, acc[j],
                    /*reuse_a=*/false, /*reuse_b=*/false);
            }
        }
        __syncthreads();                          // before buf is overwritten
    }

    // epilogue: VGPR r holds M=r (lanes 0-15) / M=r+8 (lanes 16-31), N=lane%16
#pragma unroll
    for (int j = 0; j < 8; ++j) {
#pragma unroll
        for (int r = 0; r < 8; ++r) {
            const int M   = (lane >= 16) ? (r + 8) : r;
            const int row = b0 + wave * 16 + M;
            const int col = h0 + j * 16 + lo;
            pre[(size_t)row * H + col] = acc[j][r];
        }
    }
}

// ---------------------------------------------------------------------------
// Kernel 3: per-row top-k thresholds + compaction.
// relu-after-scatter => only positive top-k survive; positive floats order as
// uint bit patterns => 3-pass (24-bit) LDS radix-select per row per code.
// Code 2 restricts candidates to dead neurons (stats > dead_threshold).
// ---------------------------------------------------------------------------
__global__ __launch_bounds__(256)
void select_topk(const float* __restrict__ pre, const int* __restrict__ stats,
                 const int* __restrict__ pk0, const int* __restrict__ pk1,
                 const int* __restrict__ pk2, const int* __restrict__ pthr,
                 int* __restrict__ idxL, float* __restrict__ valL,
                 int* __restrict__ cnts, int Bsz, int H)
{
    __shared__ unsigned hist[256];
    __shared__ unsigned sSel, sRem, sFlag, sCnt;

    const int row = blockIdx.x;
    const int tid = threadIdx.x;
    const float* __restrict__ rv = pre + (size_t)row * H;
    const int thr = *pthr;
    const int ks[3] = { *pk0, *pk1, *pk2 };

    for (int c = 0; c < 3; ++c) {
        const int  kk      = min(ks[c], LISTCAP);
        const bool useMask = (c == 2);
        unsigned prefix = 0;
        unsigned flag   = 0;

        if (tid == 0) { sRem = (unsigned)kk; sFlag = 0; }
        __syncthreads();

        for (int pass = 0; pass < 3; ++pass) {
            hist[tid] = 0;
            __syncthreads();
            if (!sFlag) {
                const int shift = 24 - 8 * pass;
                for (int h = tid; h < H; h += 256) {
                    const float v = rv[h];
                    if (v <= 0.0f) continue;
                    if (useMask && !(stats[h] > thr)) continue;
                    const unsigned u = __float_as_uint(v);
                    if (pass == 0 || (u >> (shift + 8)) == prefix)
                        atomicAdd(&hist[(u >> shift) & 255u], 1u);
                }
            }
            __syncthreads();
            if (tid == 0 && !sFlag) {
                const unsigned rem = sRem;
                unsigned cum = 0; int sel = -1;
                for (int d = 255; d >= 0; --d) {
                    if (cum + hist[d] >= rem) { sel = d; sRem = rem - cum; break; }
                    cum += hist[d];
                }
                if (sel < 0) { sFlag = 1; sSel = 0; }  // fewer than k positives
                else         { sSel = (unsigned)sel; }
            }
            __syncthreads();
            prefix = (prefix << 8) | sSel;
            flag   = sFlag;
            __syncthreads();
        }

        const unsigned thresholdBits = flag ? 1u : (prefix << 8);

        if (tid == 0) sCnt = 0;
        __syncthreads();

        int*   __restrict__ myIdx = idxL + ((size_t)c * Bsz + row) * LISTCAP;
        float* __restrict__ myVal = valL + ((size_t)c * Bsz + row) * LISTCAP;
        for (int h = tid; h < H; h += 256) {
            const float v = rv[h];
            if (v <= 0.0f) continue;
            if (useMask && !(stats[h] > thr)) continue;
            if (__float_as_uint(v) >= thresholdBits) {
                const unsigned p = atomicAdd(&sCnt, 1u);
                if ((int)p < kk) { myIdx[p] = h; myVal[p] = v; }
            }
        }
        __syncthreads();
        if (tid == 0) cnts[c * Bsz + row] = min((int)sCnt, kk);
        __syncthreads();
    }
}

// ---------------------------------------------------------------------------
// Kernel 4: sparse decoder gather. One block per (code,row).
// recon[c,b,:] = bias + sum_i val_i * W_decT[idx_i, :]
// Each thread owns a float4 of D (256*4 = 1024); W_decT rows are L2-resident.
// ---------------------------------------------------------------------------
__global__ __launch_bounds__(256)
void decode_sparse(const float* __restrict__ WdT, const float* __restrict__ bias,
                   const int* __restrict__ idxL, const float* __restrict__ valL,
                   const int* __restrict__ cnts, float* __restrict__ out,
                   int Bsz, int D)
{
    __shared__ int   sIdx[LISTCAP];
    __shared__ float sVal[LISTCAP];

    const int c   = blockIdx.x / Bsz;
    const int b   = blockIdx.x % Bsz;
    const int tid = threadIdx.x;
    const int n   = cnts[c * Bsz + b];

    const int*   __restrict__ myIdx = idxL + ((size_t)c * Bsz + b) * LISTCAP;
    const float* __restrict__ myVal = valL + ((size_t)c * Bsz + b) * LISTCAP;
    for (int i = tid; i < n; i += 256) { sIdx[i] = myIdx[i]; sVal[i] = myVal[i]; }
    __syncthreads();

    float4 acc = ((const float4*)bias)[tid];
    for (int i = 0; i < n; ++i) {
        const float  v  = sVal[i];
        const float4 w  = ((const float4*)(WdT + (size_t)sIdx[i] * D))[tid];
        if (i + 1 < n)
            __builtin_prefetch(WdT + (size_t)sIdx[i + 1] * D, 0, 0);
        acc.x += v * w.x; acc.y += v * w.y; acc.z += v * w.z; acc.w += v * w.w;
    }
    ((float4*)(out + ((size_t)c * Bsz + b) * D))[tid] = acc;
}

// ---------------------------------------------------------------------------
// Launch
// ---------------------------------------------------------------------------
extern "C" void kernel_launch(void* const* d_in, const int* in_sizes, int n_in,
                              void* d_out, int out_size, void* d_ws, size_t ws_size,
                              hipStream_t stream)
{
    const float* x     = (const float*)d_in[0];
    const float* W_enc = (const float*)d_in[1];
    const float* W_dec = (const float*)d_in[2];
    const float* bias  = (const float*)d_in[3];
    const int*   stats = (const int*)d_in[4];
    const int*   pk0   = (const int*)d_in[5];  // topk
    const int*   pk1   = (const int*)d_in[6];  // larger_topk
    const int*   pk2   = (const int*)d_in[7];  // dead_topk
    const int*   pthr  = (const int*)d_in[8];  // dead_threshold
    float* out = (float*)d_out;

    const int D   = in_sizes[3];           // 1024
    const int H   = in_sizes[1] / D;       // 16384
    const int Bsz = in_sizes[0] / D;       // 2048

    // workspace layout (floats/ints)
    float* pre  = (float*)d_ws;                          // Bsz*H
    float* WdT  = pre + (size_t)Bsz * H;                 // H*D
    float* xc   = WdT + (size_t)H * D;                   // Bsz*D
    float* valL = xc + (size_t)Bsz * D;                  // 3*Bsz*LISTCAP
    int*   idxL = (int*)(valL + (size_t)3 * Bsz * LISTCAP);
    int*   cnts = idxL + (size_t)3 * Bsz * LISTCAP;      // 3*Bsz

    const long long n4 = (long long)Bsz * D / 4;
    center_x<<<(unsigned)((n4 + 255) / 256), 256, 0, stream>>>(x, bias, xc, D, n4);
    transpose_wdec<<<dim3(H / 32, D / 32), 256, 0, stream>>>(W_dec, WdT, D, H);
    enc_gemm<<<dim3(H / BN, Bsz / BM), 256, 0, stream>>>(xc, W_enc, pre, D, H);
    select_topk<<<Bsz, 256, 0, stream>>>(pre, stats, pk0, pk1, pk2, pthr,
                                         idxL, valL, cnts, Bsz, H);
    decode_sparse<<<3 * Bsz, 256, 0, stream>>>(WdT, bias, idxL, valL, cnts,
                                               out, Bsz, D);
}